// WarpResBlock_80041010528389
// MI455X (gfx1250) — compile-verified
//
#include <hip/hip_runtime.h>
#include <cstdint>
#include <cstddef>

#define BATCH  8
#define CH_TOT 512
#define CH_X   256
#define HW     4096      // 64*64
#define WIDTH  64
#define GDIM   240

typedef __attribute__((ext_vector_type(4))) unsigned u32x4;
typedef __attribute__((ext_vector_type(8))) unsigned u32x8;

// ---------------------------------------------------------------------------
// Kernel 1: transpose x[b,c,p] (channel-major) -> xt[b,p,c] (channel contig).
// 64ch x 64px tile staged in LDS two ways (both CDNA5 DMA families):
//   rows  0..31: one TENSOR_LOAD_TO_LDS (TDM, D# descriptor, TENSORcnt);
//                D# padding (64 DW interval, 4 DW amount) produces the
//                272B (68-float) LDS row pitch directly.
//   rows 32..63: GLOBAL_LOAD_ASYNC_TO_LDS_B128 (ASYNCcnt).
// ---------------------------------------------------------------------------
__global__ __launch_bounds__(256) void xt_transpose_kernel(const float* __restrict__ data,
                                                           float* __restrict__ xt) {
    // row = channel (64), col = pixel (64); row pitch 68 floats = 272B
    // (multiple of 16B so async B128 chunks stay 16B-aligned in LDS).
    __shared__ float tile[64 * 68];

    const int blk = blockIdx.x;
    const int b   = blk >> 8;          // 256 blocks per batch
    const int ct  = (blk >> 6) & 3;    // channel tile 0..3
    const int pt  = blk & 63;          // pixel tile 0..63
    const int c0  = ct << 6;
    const int p0  = pt << 6;
    const int t   = threadIdx.x;

    const float* src = data + ((size_t)b * CH_TOT + c0) * HW + p0;

    // ---- rows 0..31 via Tensor Data Mover (issued once, by wave 0) ----
    if (t < 32) {
        const uint64_t ga   = (uint64_t)(const void*)src;            // byte addr
        const unsigned lds0 = (unsigned)(size_t)(const void*)&tile[0];
        // D# group 0: count=1 | lds_addr | global_addr[56:0] | type=2
        const u32x4 g0 = { 1u,
                           lds0,
                           (unsigned)ga,
                           ((unsigned)((ga >> 32) & 0x01FFFFFFu)) | (2u << 30) };
        // D# group 1: wg_mask=0, data_size=4B(2), pad_enable=1,
        // pad_interval=5 (64 DWORDs = 256B row), pad_amount=3 (4 DWORDs = 16B),
        // tensor_dim0=4096, tensor_dim1=32, tile_dim0=64, tile_dim1=32,
        // tile_dim2=0, tensor_dim0_stride=4096, tensor_dim1_stride=0.
        const u32x8 g1 = { (2u << 16) | (1u << 20) | (5u << 22) | (3u << 25),
                           (4096u << 16),   // tensor_dim0[15:0] @ bits 48..63
                           (32u << 16),     // tensor_dim0[31:16]=0, tensor_dim1[15:0]=32
                           (64u << 16),     // tensor_dim1[31:16]=0, tile_dim0=64
                           32u,             // tile_dim1=32, tile_dim2=0
                           4096u,           // tensor_dim0_stride[31:0]
                           0u, 0u };
        asm volatile("tensor_load_to_lds %0, %1" :: "s"(g0), "s"(g1) : "memory");
        __builtin_amdgcn_s_wait_tensorcnt(0);
    }

    // ---- rows 32..63 via async global->LDS: 512 x 16B chunks, 2/thread ----
#pragma unroll
    for (int k = 0; k < 2; ++k) {
        const int id  = (k << 8) + t;        // 0..511
        const int r   = 32 + (id >> 4);      // channel row 32..63
        const int c16 = id & 15;             // 16B chunk within the 256B row
        const float* gp = src + (size_t)r * HW + (c16 << 2);
        // Flat shared pointer's low 32 bits == LDS byte offset (ISA 10.2).
        unsigned lp = (unsigned)(size_t)(const void*)&tile[r * 68 + (c16 << 2)];
        asm volatile("global_load_async_to_lds_b128 %0, %1, off"
                     :: "v"(lp), "v"(gp) : "memory");
    }
    asm volatile("s_wait_asynccnt 0x0" ::: "memory");
    __syncthreads();

    // ---- write transposed, coalesced along the channel dimension ----
    const int cc4 = (t & 15) << 2;     // channel offset 0,4,...,60
    const int ppg = t >> 4;            // 0..15
#pragma unroll
    for (int j = 0; j < 4; ++j) {
        const int pp = ppg + (j << 4); // pixel 0..63
        float4 v;
        v.x = tile[(cc4 + 0) * 68 + pp];
        v.y = tile[(cc4 + 1) * 68 + pp];
        v.z = tile[(cc4 + 2) * 68 + pp];
        v.w = tile[(cc4 + 3) * 68 + pp];
        *(float4*)(xt + ((size_t)b * HW + p0 + pp) * CH_X + c0 + cc4) = v;
    }
}

// ---------------------------------------------------------------------------
// Per-axis bilinear tap: unnormalized coord f, tap select d in {0,1}, limit L.
// Returns weight; ci = clamped integer coord; ok = in-range mask
// (matches reference: mask computed pre-clip, gather at clipped index).
// ---------------------------------------------------------------------------
__device__ __forceinline__ float axis_tap(float f, int d, float lim, int& ci, bool& ok) {
    const float f0 = floorf(f);
    const float tc = f0 + (float)d;
    const float fr = f - f0;
    const float w  = d ? fr : (1.0f - fr);
    ok = (tc >= 0.0f) && (tc <= lim);
    ci = (int)fminf(fmaxf(tc, 0.0f), lim);
    return w;
}

// ---------------------------------------------------------------------------
// Kernel 2: fused 3-stage warp composition + sparse apply + residual.
// Block = 256 threads = 4 output pixels x 64 tap-slots/channel-groups.
// ---------------------------------------------------------------------------
__global__ __launch_bounds__(256) void warp_compose_apply_kernel(const float* __restrict__ data,
                                                                 const float* __restrict__ xt,
                                                                 float* __restrict__ out) {
    __shared__ int   sIdx[4][64];
    __shared__ float sW[4][64];

    const int t   = threadIdx.x;
    const int pix = t >> 6;            // 0..3
    const int sub = t & 63;            // 0..63
    const int b   = blockIdx.x >> 10;
    const int p   = ((blockIdx.x & 1023) << 2) + pix;
    const int oy  = p >> 6;
    const int ox  = p & 63;

    // ---- compose slot (k3,k2,k1) ----
    const int k3 = sub >> 4, k2 = (sub >> 2) & 3, k1 = sub & 3;

    // Stage 3: identity grid on 64x64 output sampling the 240x240 domain,
    // align_corners=True => coord = o * 239/63.
    const float S3 = 239.0f / 63.0f;
    int ry, rx; bool my3, mx3;
    const float wy3 = axis_tap((float)oy * S3, k3 >> 1, 239.0f, ry, my3);
    const float wx3 = axis_tap((float)ox * S3, k3 & 1,  239.0f, rx, mx3);

    const size_t bbase = (size_t)b * ((size_t)CH_TOT * HW);

    // Stage 2: g_tps lookup at stage-3 tap (channels 384..512 flat).
    const float* gt = data + bbase + (size_t)384 * HW + (size_t)(ry * GDIM + rx) * 2;
    const float fx2 = (gt[0] + 1.0f) * 0.5f * 239.0f;
    const float fy2 = (gt[1] + 1.0f) * 0.5f * 239.0f;
    int sy, sx; bool my2, mx2;
    const float wy2 = axis_tap(fy2, k2 >> 1, 239.0f, sy, my2);
    const float wx2 = axis_tap(fx2, k2 & 1,  239.0f, sx, mx2);

    // Stage 1: g_aff lookup at stage-2 tap (channels 256..384 flat).
    const float* ga = data + bbase + (size_t)256 * HW + (size_t)(sy * GDIM + sx) * 2;
    const float fx1 = (ga[0] + 1.0f) * 0.5f * 63.0f;
    const float fy1 = (ga[1] + 1.0f) * 0.5f * 63.0f;
    int ty, tx; bool my1, mx1;
    const float wy1 = axis_tap(fy1, k1 >> 1, 63.0f, ty, my1);
    const float wx1 = axis_tap(fx1, k1 & 1,  63.0f, tx, mx1);

    const bool ok = my3 && mx3 && my2 && mx2 && my1 && mx1;
    sW[pix][sub]   = ok ? (wy3 * wx3 * wy2 * wx2 * wy1 * wx1) : 0.0f;
    sIdx[pix][sub] = ty * WIDTH + tx;
    __syncthreads();

    // ---- apply: thread 'sub' owns channels 4*sub..4*sub+3 of pixel 'pix' ----
    const float4* xb = (const float4*)xt + (size_t)b * HW * (CH_X / 4);
    float4 acc = xb[(size_t)p * (CH_X / 4) + sub];   // residual: out = x + resized
#pragma unroll 4
    for (int i = 0; i < 64; ++i) {
        const float wi = sW[pix][i];                 // uniform per wave
        if (wi != 0.0f) {
            const float4 v = xb[(size_t)sIdx[pix][i] * (CH_X / 4) + sub];
            acc.x = fmaf(wi, v.x, acc.x);
            acc.y = fmaf(wi, v.y, acc.y);
            acc.z = fmaf(wi, v.z, acc.z);
            acc.w = fmaf(wi, v.w, acc.w);
        }
    }

    float* op = out + ((size_t)b * CH_X + (size_t)(sub << 2)) * HW + p;
    op[0]              = acc.x;
    op[HW]             = acc.y;
    op[2 * (size_t)HW] = acc.z;
    op[3 * (size_t)HW] = acc.w;
}

// ---------------------------------------------------------------------------
extern "C" void kernel_launch(void* const* d_in, const int* in_sizes, int n_in,
                              void* d_out, int out_size, void* d_ws, size_t ws_size,
                              hipStream_t stream) {
    const float* data = (const float*)d_in[0];   // [8,512,64,64] f32
    float* out = (float*)d_out;                  // [8,256,64,64] f32
    float* xt  = (float*)d_ws;                   // 32 MB: xt[b,p,c]

    xt_transpose_kernel<<<BATCH * 256, 256, 0, stream>>>(data, xt);
    warp_compose_apply_kernel<<<BATCH * (HW / 4), 256, 0, stream>>>(data, xt, out);
}